// PicanetL_62783831933534
// MI455X (gfx1250) — compile-verified
//
#include <hip/hip_runtime.h>
#include <hip/hip_bf16.h>

typedef float v2f __attribute__((ext_vector_type(2)));
typedef float v8f __attribute__((ext_vector_type(8)));

// Problem dims
#define BATCH 4
#define CIN   256
#define COUT1 128
#define HDIM  56
#define WDIM  56
#define HWP   3136            // 56*56
#define HPAD  68              // 56 + 2*6
#define TAPS  49

// Workspace layout (float offsets)
#define XP_OFF   0ull
#define XP_SZ    (4ull * 256 * 68 * 68)        // 4,734,976 floats
#define W1P_OFF  (XP_OFF + XP_SZ)
#define W1P_SZ   (128ull * 256 * 49)           // 1,605,632 floats
#define K1_OFF   (W1P_OFF + W1P_SZ)
#define K1_SZ    (4ull * 128 * 3136)           // 1,605,632 floats
#define ATTN_OFF (K1_OFF + K1_SZ)
#define ATTN_SZ  (4ull * 49 * 3136)            // 614,656 floats

#define ROW_STRIDE 268                     // floats per pixel row in LDS (bank-safe, 8B aligned)
#define TILE_STRIDE (16 * ROW_STRIDE + 2)  // 4290: +2 pad shifts tile1 banks by 2 (mod 4)
#define BUF_STRIDE  (2 * TILE_STRIDE)      // 8580 floats per double-buffer half

// ---------------------------------------------------------------------------
// Kernel 1: zero-pad x (B,C,56,56) -> xp (B,C,68,68)
// ---------------------------------------------------------------------------
__global__ void __launch_bounds__(256) pad_kernel(const float* __restrict__ x,
                                                  float* __restrict__ xp) {
    size_t idx = (size_t)blockIdx.x * 256 + threadIdx.x;
    if (idx >= XP_SZ) return;
    int ww = (int)(idx % HPAD);
    size_t t = idx / HPAD;
    int hh = (int)(t % HPAD);
    size_t bc = t / HPAD;                 // b*256 + c
    float v = 0.0f;
    if (hh >= 6 && hh < 62 && ww >= 6 && ww < 62)
        v = x[bc * HWP + (size_t)(hh - 6) * WDIM + (ww - 6)];
    xp[idx] = v;
}

// ---------------------------------------------------------------------------
// Kernel 2: repack W1 (O=128, I=256, 7,7) into WMMA-friendly layout:
//   W1p[((tap*64 + kk)*2 + half)*128 + oc] = float2( W1[oc][c][tap], W1[oc][c+1][tap] )
//   with c = kk*4 + half*2, tap = i*7 + j
// ---------------------------------------------------------------------------
__global__ void __launch_bounds__(256) wprep_kernel(const float* __restrict__ W1,
                                                    float* __restrict__ W1p) {
    int idx = blockIdx.x * 256 + threadIdx.x;     // over 49*64*2*128 = 802816 pairs
    if (idx >= TAPS * 64 * 2 * 128) return;
    int oc   = idx & 127;
    int r    = idx >> 7;
    int half = r & 1;
    int kk   = (r >> 1) & 63;
    int tap  = r >> 7;
    int c    = kk * 4 + half * 2;
    float a0 = W1[((size_t)oc * CIN + c) * TAPS + tap];
    float a1 = W1[((size_t)oc * CIN + c + 1) * TAPS + tap];
    W1p[2 * (size_t)idx]     = a0;
    W1p[2 * (size_t)idx + 1] = a1;
}

// ---------------------------------------------------------------------------
// Kernel 3: conv1 as implicit GEMM with V_WMMA_F32_16X16X4_F32.
// Block = 256 threads (8 waves). Block tile: 128 oc x 32 pixels.
// Each wave: one 16-oc group x two 16-pixel N-tiles (one A load feeds 2 WMMAs).
// B tiles double-buffered in LDS and filled with GLOBAL_LOAD_ASYNC_TO_LDS_B32
// (ASYNCcnt), overlapping next-tap staging with the current tap's 128 WMMAs.
// ---------------------------------------------------------------------------
__global__ void __launch_bounds__(256) conv1_wmma_kernel(const float* __restrict__ xp,
                                                         const float* __restrict__ W1p,
                                                         const float* __restrict__ b1,
                                                         float* __restrict__ k1) {
    __shared__ float xs[2 * BUF_STRIDE];   // 17160 floats = 68.6 KB (double buffer)

    const int t    = threadIdx.x;
    const int lane = t & 31;
    const int wv   = t >> 5;          // 0..7  -> oc group (16 oc each)
    const int half = lane >> 4;       // 0/1   -> K-half of fragment
    const int nl   = lane & 15;       // M row / N column within a tile

    const int ptile = blockIdx.x;     // 0..391 ; 32 consecutive flat pixels (3136%32==0)

    // Cooperative-staging mapping: thread t handles pixel px32 = t&31,
    // channels (t>>5) + 8*pass, 32 passes -> 32 async b32 copies per tap.
    const int px32 = t & 31;
    const int csub = t >> 5;          // 0..7
    const int tile = px32 >> 4;       // 0/1 -> LDS sub-tile
    const int row  = px32 & 15;

    const int p_c  = ptile * 32 + px32;
    const int b_c  = p_c / HWP;
    const int hw_c = p_c % HWP;
    const int h_c  = hw_c / WDIM;
    const int w_c  = hw_c % WDIM;

    // LDS byte address of this thread's staging slot (generic ptr low 32 bits
    // = LDS byte offset per the flat-address aperture mapping).
    const unsigned lds_slot0 =
        (unsigned)(unsigned long long)(xs + tile * TILE_STRIDE + row * ROW_STRIDE + csub);
    // Global base for this thread's (pixel, csub) at tap (ti,tj):
    const float* gbase = xp + (((size_t)(b_c * CIN + csub) * HPAD + h_c) * HPAD + w_c);

    v8f acc0 = {};
    v8f acc1 = {};

    // ---- async staging of one tap's B-tile into buffer `buf` (issue only) ----
    auto stage = [&](int tap, int buf) {
        const int ti = tap / 7;
        const int tj = tap % 7;
        const float* g = gbase + (size_t)(2 * ti) * HPAD + (2 * tj);
        unsigned lds = lds_slot0 + (unsigned)buf * (BUF_STRIDE * 4u);
        #pragma unroll
        for (int pass = 0; pass < 32; ++pass) {
            asm volatile("global_load_async_to_lds_b32 %0, %1, off"
                         :: "v"(lds), "v"(g)
                         : "memory");
            lds += 8u * 4u;                       // 8 channels per pass in LDS row
            g   += (size_t)8 * HPAD * HPAD;       // 8 channels per pass in memory
        }
    };

    // Prologue: stage tap 0 into buffer 0, then make it visible to all waves.
    stage(0, 0);
    asm volatile("s_wait_asynccnt 0x0" ::: "memory");
    __syncthreads();

    for (int tap = 0; tap < TAPS; ++tap) {
        const int cur = tap & 1;

        // Kick off next tap's staging into the back buffer (overlaps WMMAs).
        if (tap + 1 < TAPS) stage(tap + 1, 1 - cur);

        // 64 K-chunks: one A fragment feeds both N-tiles (2 WMMAs per chunk)
        const float* bufc = xs + cur * BUF_STRIDE;
        const v2f* wp   = (const v2f*)W1p + (size_t)tap * 64 * 2 * 128;
        const v2f* xsv0 = (const v2f*)(bufc + nl * ROW_STRIDE);
        const v2f* xsv1 = (const v2f*)(bufc + TILE_STRIDE + nl * ROW_STRIDE);
        #pragma unroll
        for (int kk = 0; kk < 64; ++kk) {
            v2f a   = wp[(size_t)(kk * 2 + half) * 128 + wv * 16 + nl];
            v2f b0  = xsv0[kk * 2 + half];
            v2f b1f = xsv1[kk * 2 + half];
            acc0 = __builtin_amdgcn_wmma_f32_16x16x4_f32(
                       false, a, false, b0, (short)0, acc0, false, false);
            acc1 = __builtin_amdgcn_wmma_f32_16x16x4_f32(
                       false, a, false, b1f, (short)0, acc1, false, false);
        }

        // My async copies done; barrier => back buffer complete for all waves
        // and front buffer no longer being read by anyone.
        asm volatile("s_wait_asynccnt 0x0" ::: "memory");
        __syncthreads();
    }

    // Writeout + bias. D layout: VGPR r holds M = half*8 + r, N = nl.
    const int p0  = ptile * 32 + nl;        // tile 0 pixels
    const int p1  = p0 + 16;                // tile 1 pixels
    const int bb  = p0 / HWP;               // same batch for both (32 | 3136)
    const int hw0 = p0 % HWP;
    const int hw1 = p1 % HWP;
    #pragma unroll
    for (int r = 0; r < 8; ++r) {
        int oc = wv * 16 + half * 8 + r;
        float bias = b1[oc];
        size_t base = ((size_t)(bb * COUT1 + oc)) * HWP;
        k1[base + hw0] = acc0[r] + bias;
        k1[base + hw1] = acc1[r] + bias;
    }
}

// ---------------------------------------------------------------------------
// Kernel 4: conv2 (1x1, 128 -> 49) + bias + softmax over the 49 logits.
// ---------------------------------------------------------------------------
__global__ void __launch_bounds__(256) conv2_softmax_kernel(const float* __restrict__ k1,
                                                            const float* __restrict__ W2,
                                                            const float* __restrict__ b2,
                                                            float* __restrict__ attn) {
    const int p  = blockIdx.x * 256 + threadIdx.x;   // 0..12543
    const int b  = p / HWP;
    const int hw = p % HWP;

    float l[TAPS];
    #pragma unroll
    for (int k = 0; k < TAPS; ++k) l[k] = b2[k];

    for (int c = 0; c < COUT1; ++c) {
        float v = k1[((size_t)(b * COUT1 + c)) * HWP + hw];
        #pragma unroll
        for (int k = 0; k < TAPS; ++k)
            l[k] = fmaf(v, W2[k * COUT1 + c], l[k]);
    }

    float mx = l[0];
    #pragma unroll
    for (int k = 1; k < TAPS; ++k) mx = fmaxf(mx, l[k]);
    float s = 0.0f;
    #pragma unroll
    for (int k = 0; k < TAPS; ++k) { l[k] = __expf(l[k] - mx); s += l[k]; }
    float inv = 1.0f / s;
    #pragma unroll
    for (int k = 0; k < TAPS; ++k)
        attn[((size_t)(b * TAPS + k)) * HWP + hw] = l[k] * inv;
}

// ---------------------------------------------------------------------------
// Kernel 5: out[b,c,h,w] = sum_k xp[b,c,h+2i,w+2j] * attn[b,k,h,w]
// ---------------------------------------------------------------------------
__global__ void __launch_bounds__(256) gather_kernel(const float* __restrict__ xp,
                                                     const float* __restrict__ attn,
                                                     float* __restrict__ out) {
    __shared__ float sa[TAPS * WDIM];   // 49*56 = 2744 floats
    const int rb = blockIdx.x;          // 0..223
    const int b  = rb / HDIM;
    const int h  = rb % HDIM;
    const int t  = threadIdx.x;

    for (int i = t; i < TAPS * WDIM; i += 256) {
        int k = i / WDIM, w = i % WDIM;
        sa[i] = attn[((size_t)(b * TAPS + k)) * HWP + h * WDIM + w];
    }
    __syncthreads();

    const int c  = blockIdx.y * 64 + (t >> 2);
    const int wq = t & 3;

    float o[14];
    #pragma unroll
    for (int m = 0; m < 14; ++m) o[m] = 0.0f;

    for (int k = 0; k < TAPS; ++k) {
        const int ti = k / 7, tj = k % 7;
        const float* xr = xp + (((size_t)(b * CIN + c) * HPAD + (h + 2 * ti)) * HPAD + 2 * tj);
        const float* ar = sa + k * WDIM;
        #pragma unroll
        for (int m = 0; m < 14; ++m) {
            int w = wq + 4 * m;
            o[m] = fmaf(xr[w], ar[w], o[m]);
        }
    }

    float* orow = out + ((size_t)(b * CIN + c)) * HWP + h * WDIM;
    #pragma unroll
    for (int m = 0; m < 14; ++m) orow[wq + 4 * m] = o[m];
}

// ---------------------------------------------------------------------------
extern "C" void kernel_launch(void* const* d_in, const int* in_sizes, int n_in,
                              void* d_out, int out_size, void* d_ws, size_t ws_size,
                              hipStream_t stream) {
    const float* x  = (const float*)d_in[0];
    const float* W1 = (const float*)d_in[1];
    const float* b1 = (const float*)d_in[2];
    const float* W2 = (const float*)d_in[3];
    const float* b2 = (const float*)d_in[4];
    float* out = (float*)d_out;

    float* ws   = (float*)d_ws;
    float* xp   = ws + XP_OFF;
    float* W1p  = ws + W1P_OFF;
    float* k1   = ws + K1_OFF;
    float* attn = ws + ATTN_OFF;

    // 1) zero-pad input
    pad_kernel<<<(unsigned)((XP_SZ + 255) / 256), 256, 0, stream>>>(x, xp);
    // 2) repack weights for WMMA A-fragment b64 loads
    wprep_kernel<<<(TAPS * 64 * 2 * 128 + 255) / 256, 256, 0, stream>>>(W1, W1p);
    // 3) conv1 via fp32 WMMA implicit GEMM, async double-buffered B staging
    conv1_wmma_kernel<<<392, 256, 0, stream>>>(xp, W1p, b1, k1);
    // 4) conv2 + bias + softmax -> attn
    conv2_softmax_kernel<<<49, 256, 0, stream>>>(k1, W2, b2, attn);
    // 5) weighted 49-tap gather -> out
    gather_kernel<<<dim3(224, 4), 256, 0, stream>>>(xp, attn, out);
}